// get_model_19456201851253
// MI455X (gfx1250) — compile-verified
//
#include <hip/hip_runtime.h>

#define RES    128
#define NPIX   (RES * RES)      // 16384
#define NVIEWS 6
#define BLOCK  1024             // 32 wave32s; LDS limits us to 2 WGs/WGP -> 16 waves/SIMD

// cos/sin of float32-rounded multiples of pi/2 (matches jnp f32 trig to 1st order)
#define CHP  (-4.3711390e-8f)   // cos(f32(pi/2))
#define S2HP (-8.7422777e-8f)   // sin(f32(pi))
#define C3HP ( 1.1924881e-8f)   // cos(f32(3*pi/2))

// R_v = xmat(x) @ ymat(y) @ zmat(z); transformed point = R_v @ p
// (reference applies pts @ R^T, which is the same thing)
__constant__ float ROT[NVIEWS][9] = {
    // v0: (0, 0, hp)           -> ~(-y,  x,  z)
    {  CHP, -1.f,  0.f,
       1.f,  CHP,  0.f,
       0.f,  0.f,  1.f },
    // v1: (hp, 0, hp)          -> ~(-y, -z,  x)
    {  CHP, -1.f,   0.f,
       CHP,  0.f,  -1.f,
       1.f,  CHP,   CHP },
    // v2: (2hp, 0, hp)         -> ~(-y, -x, -z)
    {  CHP,  -1.f,   0.f,
      -1.f,  -CHP, -S2HP,
      S2HP,   0.f,  -1.f },
    // v3: (3hp, 0, hp)         -> ~(-y,  z, -x)
    {  CHP,  -1.f,  0.f,
      C3HP,   0.f,  1.f,
      -1.f,  -CHP, C3HP },
    // v4: (0, -hp, hp)         -> ~(-z,  x, -y)
    {  0.f, -CHP, -1.f,
       1.f,  CHP,  0.f,
       CHP, -1.f,  CHP },
    // v5: (0, hp, hp)          -> ~( z,  x,  y)
    {  0.f, -CHP,  1.f,
       1.f,  CHP,  0.f,
      -CHP,  1.f,  CHP },
};

// v_rcp_f32 + one Newton-Raphson step (~0.5 ulp) -- replaces the ~10-instr
// precise divide sequence in the DS-bound hot loop.
__device__ __forceinline__ float fast_rcp(float d) {
    float r = __builtin_amdgcn_rcpf(d);
    float e = __builtin_fmaf(-d, r, 1.0f);
    return __builtin_fmaf(r, e, r);
}

// One workgroup owns one (batch, view) 128x128 image pair, privatized in LDS.
// 2 * 16384 * 4B = 128 KB LDS -> 2 workgroups resident per 320 KB CDNA5 WGP.
// The two 64KB arrays sit a multiple of 64 banks apart, so both ds_add_f32
// per point get the full 64-bank spread (interleaving would halve it).
__global__ __launch_bounds__(BLOCK) void depth_project_kernel(
    const float* __restrict__ pc,   // (B, 3, N) f32
    float* __restrict__ out,        // (B*6, 128, 128) f32
    int N)
{
    __shared__ float s_w[NPIX];     // scattered weights  (1/(z+eps))
    __shared__ float s_v[NPIX];     // scattered values   (z * w)

    const int bv = blockIdx.x;
    const int b  = bv / NVIEWS;
    const int v  = bv - b * NVIEWS;
    const int t  = threadIdx.x;

    // ---- zero accumulators (ds_store_b128) ----
    {
        float4 z4 = make_float4(0.f, 0.f, 0.f, 0.f);
        float4* w4 = (float4*)s_w;
        float4* v4 = (float4*)s_v;
        #pragma unroll
        for (int j = t; j < NPIX / 4; j += BLOCK) { w4[j] = z4; v4[j] = z4; }
    }
    __syncthreads();

    // ---- per-view rotation (uniform -> SGPRs) ----
    const float r0 = ROT[v][0], r1 = ROT[v][1], r2 = ROT[v][2];
    const float r3 = ROT[v][3], r4 = ROT[v][4], r5 = ROT[v][5];
    const float r6 = ROT[v][6], r7 = ROT[v][7], r8 = ROT[v][8];

    const float* base = pc + (size_t)b * 3 * (size_t)N;
    const float4* px4 = (const float4*)(base);
    const float4* py4 = (const float4*)(base + N);
    const float4* pz4 = (const float4*)(base + 2 * N);
    const int n4 = N >> 2;

    auto project_one = [&](float x, float y, float z) {
        float tx = r0 * x + r1 * y + r2 * z;
        float ty = r3 * x + r4 * y + r5 * z;
        float tz = r6 * x + r7 * y + r8 * z + 1.4f;   // minus translation (0,0,-1.4)
        float inv = fast_rcp(tz + 1e-12f);            // shared by projection & weight
        float ex = ceilf(__builtin_fmaf(tx * inv, (float)(RES / 2), (float)(RES / 2) - 0.5f));
        float ey = ceilf(__builtin_fmaf(ty * inv, (float)(RES / 2), (float)(RES / 2) - 0.5f));
        // masked-out points contribute exactly 0 in the reference -> skip them
        if (ex >= 0.0f && ex <= (float)(RES - 1) &&
            ey >= 0.0f && ey <= (float)(RES - 1) && tz >= 0.0f) {
            int idx = (int)ex * RES + (int)ey;
            atomicAdd(&s_w[idx], inv);        // ds_add_f32 (no return)
            atomicAdd(&s_v[idx], tz * inv);   // ds_add_f32 (no return)
        }
    };

    // ---- stream points: coalesced global_load_b128 per plane + prefetch ----
    for (int i = t; i < n4; i += BLOCK) {
        if (i + BLOCK < n4) {
            __builtin_prefetch(&px4[i + BLOCK], 0, 3);
            __builtin_prefetch(&py4[i + BLOCK], 0, 3);
            __builtin_prefetch(&pz4[i + BLOCK], 0, 3);
        }
        float4 X = px4[i];
        float4 Y = py4[i];
        float4 Z = pz4[i];
        project_one(X.x, Y.x, Z.x);
        project_one(X.y, Y.y, Z.y);
        project_one(X.z, Y.z, Z.z);
        project_one(X.w, Y.w, Z.w);
    }
    // scalar tail (N not a multiple of 4 -- not hit for the reference shapes)
    for (int i = (n4 << 2) + t; i < N; i += BLOCK) {
        project_one(base[i], base[N + i], base[2 * N + i]);
    }

    __syncthreads();

    // ---- normalize and write finished tile (ds_load_b128 + global_store_b128) ----
    {
        float4* w4 = (float4*)s_w;
        float4* v4 = (float4*)s_v;
        float4* o4 = (float4*)(out + (size_t)bv * NPIX);
        #pragma unroll
        for (int j = t; j < NPIX / 4; j += BLOCK) {
            float4 w = w4[j];
            float4 a = v4[j];
            w.x += (w.x == 0.0f) ? 1.0f : 0.0f;
            w.y += (w.y == 0.0f) ? 1.0f : 0.0f;
            w.z += (w.z == 0.0f) ? 1.0f : 0.0f;
            w.w += (w.w == 0.0f) ? 1.0f : 0.0f;
            float4 d;
            d.x = a.x / w.x;    // 16384 exact divides per WG: negligible, keep precise
            d.y = a.y / w.y;
            d.z = a.z / w.z;
            d.w = a.w / w.w;
            o4[j] = d;
        }
    }
}

extern "C" void kernel_launch(void* const* d_in, const int* in_sizes, int n_in,
                              void* d_out, int out_size, void* d_ws, size_t ws_size,
                              hipStream_t stream) {
    (void)n_in; (void)d_ws; (void)ws_size;
    const float* pc = (const float*)d_in[0];
    float* out = (float*)d_out;

    int B = out_size / (NVIEWS * NPIX);                 // 32 for the reference shapes
    if (B <= 0) B = 1;
    int N = (int)((long long)in_sizes[0] / (3LL * B));  // 131072

    dim3 grid(B * NVIEWS);                              // 192 workgroups, one per (b, view)
    depth_project_kernel<<<grid, BLOCK, 0, stream>>>(pc, out, N);
}